// ReprojectionLayer_11209864643114
// MI455X (gfx1250) — compile-verified
//
#include <hip/hip_runtime.h>

// Problem constants (match reference)
#define B_  4
#define C_  12
#define K_  23
#define H_  256
#define W_  320
#define HW_ (H_ * W_)          // 81920
#define L_  192                // lookup grid dim
#define G_  64                 // crop cube dim
#define G3_ (G_ * G_ * G_)     // 262144
#define SPACING_ 2.0f

typedef unsigned int u32x4 __attribute__((ext_vector_type(4)));
typedef int          i32x8 __attribute__((ext_vector_type(8)));
typedef int          i32x4 __attribute__((ext_vector_type(4)));

// Block = (b, x, y0..y0+3) ; 256 threads = 4 y-rows x 64 z
__global__ __launch_bounds__(256) void reproject_gather_kernel(
    const float* __restrict__ heatmaps,  // [B, C, K, H, W]
    const float* __restrict__ center,    // [B, 3]
    const int*   __restrict__ lookup,    // [C, L, L, L]
    float*       __restrict__ out)       // [B, K, G, G, G]
{
    // 12 cameras * 4 y-rows * 64 z = 3072 ints = 12 KB, filled by the TDM
    __shared__ int sidx[C_ * 4 * G_];

    const int bx  = blockIdx.x;
    const int b   = bx >> 10;            // / (64 * 16)
    const int rem = bx & 1023;
    const int x   = rem >> 4;            // 0..63
    const int y0  = (rem & 15) << 2;     // 0,4,...,60

    // Uniform crop start (truncation toward zero, like torch .int())
    const int sx = (int)(center[3 * b + 0] * (1.0f / SPACING_)) - G_ / 2;
    const int sy = (int)(center[3 * b + 1] * (1.0f / SPACING_)) - G_ / 2;
    const int sz = (int)(center[3 * b + 2] * (1.0f / SPACING_)) - G_ / 2;

    // ---- Tensor Data Mover: DMA the strided 3D lookup tile into LDS ----
    // Tile: dim0 = 64 contiguous z, dim1 = 4 y-rows (stride 192),
    //       dim2 = 12 cameras (stride 192^3 via tensor_dim1_stride).
    if (threadIdx.x == 0) {
        unsigned long long gaddr =
            (unsigned long long)(const void*)lookup +
            4ull * ((unsigned long long)(unsigned)(sx + x) * (L_ * L_) +
                    (unsigned long long)(unsigned)(sy + y0) * L_ +
                    (unsigned long long)(unsigned)sz);
        // Generic shared pointer: low 32 bits are the LDS byte offset.
        unsigned int ldsoff = (unsigned int)(unsigned long long)(&sidx[0]);

        // D# group 0: count=1 | lds_addr | global_addr[56:0] | type=2
        u32x4 g0;
        g0[0] = 1u;                                   // count = 1 (valid)
        g0[1] = ldsoff;                               // lds_addr
        g0[2] = (unsigned int)gaddr;                  // global_addr[31:0]
        g0[3] = ((unsigned int)(gaddr >> 32) & 0x01FFFFFFu) | (2u << 30); // addr[56:32] | type=2

        // D# group 1
        i32x8 g1;
        g1[0] = (2 << 16);                            // data_size = 4 bytes; no multicast/pad/iterate
        g1[1] = (int)((192u & 0xFFFFu) << 16);        // tensor_dim0[15:0] in bits 31:16
        g1[2] = (int)((192u >> 16) | (192u << 16));   // tensor_dim0[31:16] | tensor_dim1[15:0]
        g1[3] = (int)((192u >> 16) | (64u  << 16));   // tensor_dim1[31:16] | tile_dim0 = 64
        g1[4] = (int)(4u | (12u << 16));              // tile_dim1 = 4 | tile_dim2 = 12
        g1[5] = 192;                                  // tensor_dim0_stride[31:0] (y step)
        g1[6] = (int)(((7077888u & 0xFFFFu) << 16));  // dim0_stride[47:32]=0 | dim1_stride[15:0]
        g1[7] = (int)(7077888u >> 16);                // tensor_dim1_stride[47:16] (c step = 192^3)

        // D# group 2: tensor_dim2 = 12 (camera count), rest unused
        i32x4 g2; g2[0] = 12; g2[1] = 0; g2[2] = 0; g2[3] = 0;
        i32x4 g3; g3[0] = 0;  g3[1] = 0; g3[2] = 0; g3[3] = 0;
        // Extra group required by the 6-arg (clang-23) builtin variant; zero-filled.
        i32x8 g4; g4[0] = 0; g4[1] = 0; g4[2] = 0; g4[3] = 0;
                  g4[4] = 0; g4[5] = 0; g4[6] = 0; g4[7] = 0;

        __builtin_amdgcn_tensor_load_to_lds(g0, g1, g2, g3, g4, 0);
        __builtin_amdgcn_s_wait_tensorcnt(0);         // TENSORcnt -> 0: tile is in LDS
    }
    __syncthreads();

    const int z  = threadIdx.x & 63;
    const int dy = threadIdx.x >> 6;

    float acc[K_];
#pragma unroll
    for (int k = 0; k < K_; ++k) acc[k] = 0.0f;

    const float* hb = heatmaps + (size_t)b * (C_ * K_ * (size_t)HW_);

#pragma unroll 1
    for (int c = 0; c < C_; ++c) {
        const int idx = sidx[(c * 4 + dy) * G_ + z];       // conflict-free LDS read
        const float* hk = hb + (size_t)c * (K_ * (size_t)HW_);
        // 23 independent L2 gathers in flight per camera
#pragma unroll
        for (int k = 0; k < K_; ++k)
            acc[k] += hk[(size_t)k * HW_ + (size_t)idx];
    }

    const size_t voxel = (size_t)x * (G_ * G_) + (size_t)(y0 + dy) * G_ + (size_t)z;
    float* ob = out + ((size_t)b * K_) * (size_t)G3_ + voxel;
#pragma unroll
    for (int k = 0; k < K_; ++k)
        ob[(size_t)k * G3_] = acc[k] * (1.0f / 12.0f);     // mean over cameras
}

extern "C" void kernel_launch(void* const* d_in, const int* in_sizes, int n_in,
                              void* d_out, int out_size, void* d_ws, size_t ws_size,
                              hipStream_t stream) {
    (void)in_sizes; (void)n_in; (void)out_size; (void)d_ws; (void)ws_size;
    const float* heatmaps = (const float*)d_in[0];   // [4,12,23,256,320] f32
    const float* center   = (const float*)d_in[1];   // [4,3] f32
    const int*   lookup   = (const int*)d_in[2];     // [12,192,192,192] i32
    float*       out      = (float*)d_out;           // [4,23,64,64,64] f32

    dim3 grid(B_ * G_ * (G_ / 4));                   // 4 * 64 * 16 = 4096 blocks
    dim3 block(256);
    reproject_gather_kernel<<<grid, block, 0, stream>>>(heatmaps, center, lookup, out);
}